// VectorQuantizer_19464791785678
// MI455X (gfx1250) — compile-verified
//
#include <hip/hip_runtime.h>

typedef float v2f __attribute__((ext_vector_type(2)));
typedef float v8f __attribute__((ext_vector_type(8)));

#define K_CODES 512
#define DIM 64
#define WROW 68                 // padded LDS row stride (floats): 4n mod 64 bank walk, conflict-free b64 reads
#define ROWS_PER_BLOCK 128      // 8 waves * 16 rows
#define NUM_ROWS 131072
#define NUM_BLOCKS (NUM_ROWS / ROWS_PER_BLOCK)   // 1024
#define OUT_ELEMS (NUM_ROWS * DIM)               // 8388608

#if defined(__has_builtin)
#if __has_builtin(__builtin_amdgcn_sched_group_barrier)
#define HAVE_SGB 1
#endif
#endif

__global__ __launch_bounds__(256) void vq_main(const float* __restrict__ latents,
                                               const float* __restrict__ weight,
                                               float* __restrict__ out,
                                               float* __restrict__ partial) {
  extern __shared__ float smem[];
  float* lds_w   = smem;                                  // 512*68 floats
  float* lds_w2  = lds_w + K_CODES * WROW;                // 512 floats
  float* lds_z2  = lds_w2 + K_CODES;                      // 128 floats
  int*   lds_bn  = (int*)(lds_z2 + ROWS_PER_BLOCK);       // 128 ints
  float* lds_red = (float*)(lds_bn + ROWS_PER_BLOCK);     // 256 floats

  const int tid   = threadIdx.x;
  const int wave  = tid >> 5;
  const int lane  = tid & 31;
  const int nlane = lane & 15;
  const int koff  = (lane >> 4) * 2;       // half-wave K offset (A & B fragment layout)
  const int row0  = blockIdx.x * ROWS_PER_BLOCK + wave * 16;

  // ---- stage codebook into padded LDS ----
  for (int i = tid; i < K_CODES * DIM / 4; i += 256) {
    int n  = i >> 4;
    int d4 = (i & 15) * 4;
    const float4 w = *(const float4*)(weight + n * DIM + d4);
    float* dst = lds_w + n * WROW + d4;
    dst[0] = w.x; dst[1] = w.y; dst[2] = w.z; dst[3] = w.w;
  }
  __syncthreads();

  // ---- |w|^2 per code (from LDS) ----
  for (int n = tid; n < K_CODES; n += 256) {
    float s = 0.f;
    const float* wr = lds_w + n * WROW;
    #pragma unroll
    for (int d = 0; d < DIM; ++d) s = fmaf(wr[d], wr[d], s);
    lds_w2[n] = s;
  }

  // ---- preload A fragments (16x4 f32 WMMA layout) and |z|^2 per row ----
  const float* zrow = latents + (size_t)(row0 + nlane) * DIM + koff;
  v2f a[16];
  float zp = 0.f;
  #pragma unroll
  for (int k = 0; k < 16; ++k) {
    a[k] = *(const v2f*)(zrow + k * 4);
    zp = fmaf(a[k].x, a[k].x, zp);
    zp = fmaf(a[k].y, a[k].y, zp);
  }
  zp += __shfl_xor(zp, 16, 32);            // lanes L and L+16 together hold the full row
  if (lane < 16) lds_z2[wave * 16 + nlane] = zp;
  __syncthreads();

  float z2r[8];
  #pragma unroll
  for (int r = 0; r < 8; ++r)
    z2r[r] = lds_z2[wave * 16 + r + ((lane >> 4) << 3)];  // C/D layout: M = r (+8 for upper half)

  // ---- distance argmin over 32 N-tiles of 16 codes, K=64 via 16 x WMMA_F32_16X16X4 ----
  // Double-buffered B fragments + sched_group_barrier pipelining: tile t+1's 8 DS reads are
  // interleaved into tile t's 16-deep WMMA chain so the chain hides all LDS latency.
  float bestD[8]; int bestN[8];
  #pragma unroll
  for (int r = 0; r < 8; ++r) { bestD[r] = 3.402823e38f; bestN[r] = 0; }

  v2f bb[2][16];
  {
    const float* bp = lds_w + nlane * WROW + koff;       // tile 0
    #pragma unroll
    for (int k = 0; k < 16; ++k) bb[0][k] = *(const v2f*)(bp + k * 4);
  }

  #pragma unroll 2
  for (int t = 0; t < 32; ++t) {
    const int cur = t & 1;
    // prefetch next tile's B fragments (wraps to tile 0 on the last iter; harmless)
    {
      const int tn = (t + 1) & 31;
      const float* bp = lds_w + (tn * 16 + nlane) * WROW + koff;
      #pragma unroll
      for (int k = 0; k < 16; ++k) bb[cur ^ 1][k] = *(const v2f*)(bp + k * 4);
    }

    v8f c = {0.f, 0.f, 0.f, 0.f, 0.f, 0.f, 0.f, 0.f};
    #pragma unroll
    for (int k = 0; k < 16; ++k)
      c = __builtin_amdgcn_wmma_f32_16x16x4_f32(false, a[k], false, bb[cur][k],
                                                (short)0, c, false, false);

#ifdef HAVE_SGB
    // Desired steady-state schedule per tile: (1 DS-read, 2 WMMA) x 8.
    // DS reads belong to tile t+1 (double buffer); WMMAs consume tile t.
    #pragma unroll
    for (int g = 0; g < 8; ++g) {
      __builtin_amdgcn_sched_group_barrier(0x100, 1, 0);  // 1 DS read
      __builtin_amdgcn_sched_group_barrier(0x008, 2, 0);  // 2 WMMA
    }
#endif

    const int   n   = t * 16 + nlane;
    const float w2n = lds_w2[n];
    #pragma unroll
    for (int r = 0; r < 8; ++r) {
      float score = (z2r[r] - 2.0f * c[r]) + w2n;
      if (score < bestD[r]) { bestD[r] = score; bestN[r] = n; }
    }
  }

  // ---- argmin reduce within each 16-lane half (masks <16 stay in-half), tie -> smaller index ----
  #pragma unroll
  for (int r = 0; r < 8; ++r) {
    float d = bestD[r]; int n = bestN[r];
    #pragma unroll
    for (int m = 1; m <= 8; m <<= 1) {
      float od = __shfl_xor(d, m, 32);
      int   on = __shfl_xor(n, m, 32);
      if (od < d || (od == d && on < n)) { d = od; n = on; }
    }
    if (lane == 0)  lds_bn[wave * 16 + r]     = n;   // M = r
    if (lane == 16) lds_bn[wave * 16 + 8 + r] = n;   // M = r + 8
  }
  __syncthreads();

  // ---- gather quantized, straight-through output, loss partial ----
  const float* zb = latents + (size_t)row0 * DIM;
  float*       ob = out     + (size_t)row0 * DIM;
  float acc = 0.f;
  #pragma unroll
  for (int i = 0; i < 8; ++i) {
    int f4 = i * 32 + lane;
    int m  = f4 >> 4;
    int d4 = (f4 & 15) * 4;
    int bn = lds_bn[wave * 16 + m];
    const float* q = lds_w + bn * WROW + d4;
    const float* z = zb + m * DIM + d4;
    float o[4];
    #pragma unroll
    for (int j = 0; j < 4; ++j) {
      float zz = z[j], qq = q[j];
      float df = zz - qq;
      acc = fmaf(df, df, acc);
      o[j] = zz + (qq - zz);               // straight-through: matches reference fp ordering
    }
    *(float4*)(ob + m * DIM + d4) = make_float4(o[0], o[1], o[2], o[3]);
  }

  lds_red[tid] = acc;
  __syncthreads();
  for (int s = 128; s > 0; s >>= 1) {
    if (tid < s) lds_red[tid] += lds_red[tid + s];
    __syncthreads();
  }
  if (tid == 0) partial[blockIdx.x] = lds_red[0];
}

__global__ __launch_bounds__(256) void vq_loss_reduce(const float* __restrict__ partial,
                                                      float* __restrict__ out) {
  __shared__ float red[256];
  int tid = threadIdx.x;
  red[tid] = partial[tid] + partial[tid + 256] + partial[tid + 512] + partial[tid + 768];
  __syncthreads();
  for (int k = 128; k > 0; k >>= 1) {
    if (tid < k) red[tid] += red[tid + k];
    __syncthreads();
  }
  if (tid == 0) {
    float m = red[0] / (float)OUT_ELEMS;   // mean((z-q)^2)
    out[OUT_ELEMS] = m + 0.25f * m;        // codebook + COMMITMENT_COST * commitment
  }
}

extern "C" void kernel_launch(void* const* d_in, const int* in_sizes, int n_in,
                              void* d_out, int out_size, void* d_ws, size_t ws_size,
                              hipStream_t stream) {
  const float* latents = (const float*)d_in[0];   // [8192,1024]
  const float* weight  = (const float*)d_in[1];   // [512,64]
  float* out     = (float*)d_out;                 // 8388608 outputs + 1 loss
  float* partial = (float*)d_ws;                  // 1024 block partials

  size_t smem = (size_t)(K_CODES * WROW + K_CODES) * sizeof(float)   // lds_w + lds_w2
              + (size_t)ROWS_PER_BLOCK * sizeof(float)               // lds_z2
              + (size_t)ROWS_PER_BLOCK * sizeof(int)                 // lds_bn
              + 256 * sizeof(float);                                 // lds_red

  vq_main<<<NUM_BLOCKS, 256, smem, stream>>>(latents, weight, out, partial);
  vq_loss_reduce<<<1, 256, 0, stream>>>(partial, out);
}